// NTM_4148938408114
// MI455X (gfx1250) — compile-verified
//
#include <hip/hip_runtime.h>
#include <hip/hip_bf16.h>
#include <math.h>

// ---------------------------------------------------------------------------
// NTM forward for MI455X (gfx1250, wave32).
// Bandwidth-bound: ~48MB of traffic per timestep; 16MB memory state stays
// L2-resident (192MB L2). fp32 WMMA (V_WMMA_F32_16X16X4_F32) for the
// controller GEMM and the mem-vs-key cosine-similarity einsum.
// Mem tiles staged to LDS with GLOBAL_LOAD_ASYNC_TO_LDS_B128 (ASYNCcnt).
// ---------------------------------------------------------------------------

typedef __attribute__((ext_vector_type(2))) float v2f;
typedef __attribute__((ext_vector_type(8))) float v8f;

#define T_     32
#define B_     16
#define IN_    64
#define OUT_   64
#define W_     64
#define L_     4096
#define RS_    70     // READ_SIZE  = W+3+S
#define CIN_   192    // IN+OUT+W
#define COUT_  332    // OUT + READ_SIZE + WRITE_SIZE
#define LCHUNK 256
#define NBLK   16     // L_/LCHUNK

// workspace layout (float offsets)
#define OFF_MEM  0
#define SZ_MEM   (B_*L_*W_)                 // 4,194,304 floats (16MB)
#define OFF_U    (OFF_MEM + SZ_MEM)
#define SZ_U     (B_*2*L_)
#define OFF_WB   (OFF_U + SZ_U)
#define OFF_RWS  (OFF_WB + SZ_U)
#define OFF_WWS  (OFF_RWS + B_*L_)
#define OFF_RST  (OFF_WWS + B_*L_)
#define OFF_OST  (OFF_RST + B_*W_)
#define OFF_CO   (OFF_OST + B_*OUT_)
#define OFF_HP   (OFF_CO + B_*COUT_)        // per (b,head): beta,g,gamma,s0,s1,s2,inv_nk,pad
#define OFF_USUM (OFF_HP + B_*2*8)
#define OFF_WSUM (OFF_USUM + B_*2)
#define OFF_PSU  (OFF_WSUM + B_*2)
#define OFF_PSW  (OFF_PSU + B_*2*NBLK)
#define OFF_PSR  (OFF_PSW + B_*2*NBLK)
#define WS_NEED_FLOATS (OFF_PSR + B_*NBLK*W_)

__device__ __forceinline__ float sigmoid_f(float x) {
    return 1.0f / (1.0f + __expf(-x));
}
__device__ __forceinline__ float softplus_f(float x) {
    return (x > 20.0f) ? x : log1pf(__expf(x));
}

// memory -> LDS without VGPR round-trip; tracked by ASYNCcnt (gfx1250).
__device__ __forceinline__ void async_load_lds_b128(void* lds_dst, const void* gsrc) {
    unsigned lds_addr = (unsigned)(size_t)lds_dst;          // low 32 bits = LDS offset
    unsigned long long ga = (unsigned long long)(size_t)gsrc;
    asm volatile("global_load_async_to_lds_b128 %0, %1, off"
                 :: "v"(lds_addr), "v"(ga) : "memory");
}
__device__ __forceinline__ void wait_asynccnt0() {
    asm volatile("s_wait_asynccnt 0" ::: "memory");
}

// ---------------------------------------------------------------------------
// init: zero mem / r / out, delta-initialize read & write weights
// ---------------------------------------------------------------------------
__global__ __launch_bounds__(256) void ntm_init(float* __restrict__ ws) {
    size_t idx = (size_t)blockIdx.x * 256 + threadIdx.x;
    if (idx < (size_t)(SZ_MEM / 4)) {
        float4 z = {0.f, 0.f, 0.f, 0.f};
        ((float4*)(ws + OFF_MEM))[idx] = z;
    }
    if (idx < (size_t)(B_ * L_)) {
        float v = ((idx & (L_ - 1)) == 0) ? 1.0f : 0.0f;
        ws[OFF_RWS + idx] = v;
        ws[OFF_WWS + idx] = v;
    }
    if (idx < (size_t)(B_ * W_)) {
        ws[OFF_RST + idx] = 0.f;
        ws[OFF_OST + idx] = 0.f;
    }
}

// ---------------------------------------------------------------------------
// controller: co = [x_t | out | r] @ Wc + bc  via fp32 WMMA, then head params
// 1 block, 768 threads (24 waves). Waves 0..20 each own one 16-col tile of co.
// ---------------------------------------------------------------------------
__global__ __launch_bounds__(768)
void ntm_controller(const float* __restrict__ x_t, const float* __restrict__ Wc,
                    const float* __restrict__ bc, float* __restrict__ ws,
                    float* __restrict__ dout, int t)
{
    __shared__ float cat[B_][CIN_ + 1];   // +1 pad: conflict-free A-frag reads
    __shared__ float coS[B_ * COUT_];
    float* co        = ws + OFF_CO;
    float* ost       = ws + OFF_OST;
    const float* rst = ws + OFF_RST;
    float* hp        = ws + OFF_HP;
    int tid = threadIdx.x;

    // stage A: assemble concat input in LDS
    for (int idx = tid; idx < B_ * CIN_; idx += 768) {
        int b = idx / CIN_, i = idx % CIN_;
        float v;
        if (i < IN_)             v = x_t[b * IN_ + i];
        else if (i < IN_ + OUT_) v = ost[b * OUT_ + (i - IN_)];
        else                     v = rst[b * W_ + (i - IN_ - OUT_)];
        cat[b][i] = v;
    }
    __syncthreads();

    int wv = tid >> 5, lane = tid & 31;
    if (wv < 21) {
        int n = wv * 16 + (lane & 15);
        int nld = (n < COUT_) ? n : (COUT_ - 1);  // clamp: branch-free K loop
        int m = lane & 15;
        int khalf = (lane >> 4) * 2;  // A/B 32-bit frag: vgpr v holds K = v + 2*(lane/16)
        v8f acc = {0.f, 0.f, 0.f, 0.f, 0.f, 0.f, 0.f, 0.f};
        for (int c = 0; c < CIN_ / 4; ++c) {
            int kk = c * 4 + khalf;
            v2f a;  a.x = cat[m][kk];            a.y = cat[m][kk + 1];
            v2f bb; bb.x = Wc[kk * COUT_ + nld]; bb.y = Wc[(kk + 1) * COUT_ + nld];
            acc = __builtin_amdgcn_wmma_f32_16x16x4_f32(false, a, false, bb,
                                                        (short)0, acc, false, false);
        }
        if (n < COUT_) {
            float bias = bc[n];
            for (int v = 0; v < 8; ++v) {
                int mm = v + ((lane >> 4) << 3);      // D: vgpr v -> M = v + 8*(lane/16)
                float val = acc[v] + bias;
                coS[mm * COUT_ + n] = val;
                co[mm * COUT_ + n]  = val;
            }
        }
    }
    __syncthreads();

    // stage C1: derived head params (beta, g, gamma, softmax(s), 1/||k||)
    for (int idx = wv; idx < B_ * 2; idx += 24) {
        int b = idx >> 1, h = idx & 1;
        const float* base = &coS[b * COUT_ + OUT_ + h * RS_];
        float k0 = base[2 * lane], k1 = base[2 * lane + 1];
        float ss = k0 * k0 + k1 * k1;
        for (int off = 16; off >= 1; off >>= 1) ss += __shfl_down(ss, off);
        if (lane == 0) {
            float beta  = softplus_f(base[64]);
            float g     = sigmoid_f(base[65]);
            float gamma = softplus_f(base[66]) + 1.0f;
            float s0 = base[67], s1 = base[68], s2 = base[69];
            float mx = fmaxf(s0, fmaxf(s1, s2));
            float e0 = __expf(s0 - mx), e1 = __expf(s1 - mx), e2 = __expf(s2 - mx);
            float inv = 1.0f / (e0 + e1 + e2);
            float* H = &hp[idx * 8];
            H[0] = beta; H[1] = g; H[2] = gamma;
            H[3] = e0 * inv; H[4] = e1 * inv; H[5] = e2 * inv;
            H[6] = rsqrtf(ss + 1e-14f);
        }
    }
    // stage C2: emit out_t (also new recurrent out state)
    for (int idx = tid; idx < B_ * OUT_; idx += 768) {
        float v = coS[(idx / OUT_) * COUT_ + (idx % OUT_)];
        ost[idx] = v;
        dout[(size_t)t * B_ * OUT_ + idx] = v;
    }
}

// ---------------------------------------------------------------------------
// score: u = exp(beta*(cos(mem_l,k)-1)) for both heads, block-partial sums.
// Stable softmax without a global max pass (cos <= 1). One pass over mem,
// staged to LDS via GLOBAL_LOAD_ASYNC_TO_LDS_B128.
// grid = B*NBLK blocks, 256 threads (8 waves); wave handles 2 16-row tiles.
// ---------------------------------------------------------------------------
__global__ __launch_bounds__(256)
void ntm_score(float* __restrict__ ws)
{
    __shared__ __align__(16) float tileS[8][16][68];  // 68-stride: 16B rows, few conflicts
    __shared__ float normS[8][16];
    __shared__ float wsumS[8][2];
    const float* mem = ws + OFF_MEM;
    const float* co  = ws + OFF_CO;
    const float* hp  = ws + OFF_HP;
    float* u   = ws + OFF_U;
    float* psu = ws + OFF_PSU;

    int blk = blockIdx.x;
    int b = blk / NBLK, lb = blk % NBLK;
    int l0 = lb * LCHUNK;
    int tid = threadIdx.x, wv = tid >> 5, lane = tid & 31;
    int n = lane & 15;                 // WMMA column = head (0=read,1=write)
    int khalf = (lane >> 4) * 2;

    // B-fragments: keys pre-scaled by beta/||k|| so D = beta*cos*||mem_row||
    float bfx[16], bfy[16];
    float beta_n = 0.f;
    if (n < 2) {
        const float* kp = &co[b * COUT_ + OUT_ + n * RS_];
        const float* H  = &hp[(b * 2 + n) * 8];
        beta_n = H[0];
        float scale = H[0] * H[6];
        for (int c = 0; c < 16; ++c) {
            int kk = c * 4 + khalf;
            bfx[c] = kp[kk] * scale;
            bfy[c] = kp[kk + 1] * scale;
        }
    } else {
        for (int c = 0; c < 16; ++c) { bfx[c] = 0.f; bfy[c] = 0.f; }
    }

    float hsum = 0.f;  // this lane's partial of sum(u) for its head column
    for (int tix = 0; tix < 2; ++tix) {
        int lt0 = l0 + (wv * 2 + tix) * 16;
        // async-stage 16x64 tile to LDS (no VGPR round trip; ASYNCcnt tracked)
        for (int i = 0; i < 8; ++i) {
            int flat = i * 32 + lane;
            int row = flat >> 4, c4 = (flat & 15) * 4;
            async_load_lds_b128(&tileS[wv][row][c4],
                                &mem[((size_t)(b * L_ + lt0 + row)) * W_ + c4]);
        }
        wait_asynccnt0();
        v8f acc = {0.f, 0.f, 0.f, 0.f, 0.f, 0.f, 0.f, 0.f};
        int m = lane & 15;
        float ssq = 0.f;
        for (int c = 0; c < 16; ++c) {
            int kk = c * 4 + khalf;
            v2f a;  a.x = tileS[wv][m][kk];  a.y = tileS[wv][m][kk + 1];
            ssq += a.x * a.x + a.y * a.y;          // row-norm from A fragments
            v2f bb; bb.x = bfx[c]; bb.y = bfy[c];
            acc = __builtin_amdgcn_wmma_f32_16x16x4_f32(false, a, false, bb,
                                                        (short)0, acc, false, false);
        }
        // combine the two half-row partials (lane m and lane m+16): wave32 xor-16
        float nv = sqrtf(ssq + __shfl_xor(ssq, 16) + 1e-14f);
        if (lane < 16) normS[wv][lane] = nv;
        if (n < 2) {
            for (int v = 0; v < 8; ++v) {
                int mm = v + ((lane >> 4) << 3);
                float cosb = acc[v] / normS[wv][mm];      // = beta * cosine
                float uv = __expf(cosb - beta_n);         // shift-invariant softmax
                u[((size_t)(b * 2 + n)) * L_ + lt0 + mm] = uv;
                hsum += uv;
            }
        }
    }
    float hx = hsum + __shfl_xor(hsum, 16);
    if (lane < 2) wsumS[wv][lane] = hx;
    __syncthreads();
    if (tid < 2) {
        float s = 0.f;
        for (int w = 0; w < 8; ++w) s += wsumS[w][tid];
        psu[(b * 2 + tid) * NBLK + lb] = s;    // deterministic block partials
    }
}

// ---------------------------------------------------------------------------
// 32-thread reduction: per (b,head) sum of NBLK partials -> reciprocal
// ---------------------------------------------------------------------------
__global__ __launch_bounds__(32)
void ntm_reduce32(const float* __restrict__ ps, float* __restrict__ inv_out)
{
    int i = threadIdx.x;
    float s = 0.f;
    for (int j = 0; j < NBLK; ++j) s += ps[i * NBLK + j];
    inv_out[i] = 1.0f / s;
}

// ---------------------------------------------------------------------------
// interpolate + circular shift conv + sharpen; block-partial sums of w
// grid = B*2*NBLK blocks of 256 (one l per thread)
// ---------------------------------------------------------------------------
__global__ __launch_bounds__(256)
void ntm_shift(float* __restrict__ ws)
{
    __shared__ float red[256];
    const float* u    = ws + OFF_U;
    const float* us   = ws + OFF_USUM;
    const float* hp   = ws + OFF_HP;
    const float* rws  = ws + OFF_RWS;
    const float* wws  = ws + OFF_WWS;
    float* wb  = ws + OFF_WB;
    float* psw = ws + OFF_PSW;

    int blk = blockIdx.x;
    int bh = blk / NBLK, sub = blk % NBLK;
    int l = sub * LCHUNK + threadIdx.x;
    const float* up = &u[(size_t)bh * L_];
    const float* wp = (bh & 1) ? &wws[(size_t)(bh >> 1) * L_]
                               : &rws[(size_t)(bh >> 1) * L_];
    const float* H = &hp[bh * 8];
    float g = H[1], gamma = H[2], s0 = H[3], s1 = H[4], s2 = H[5];
    float uinv = us[bh];
    int lm = (l + L_ - 1) & (L_ - 1);
    int lp = (l + 1) & (L_ - 1);
    float wg_m = g * up[lm] * uinv + (1.f - g) * wp[lm];
    float wg_0 = g * up[l]  * uinv + (1.f - g) * wp[l];
    float wg_p = g * up[lp] * uinv + (1.f - g) * wp[lp];
    float wt = s0 * wg_m + s1 * wg_0 + s2 * wg_p;   // roll(+1),identity,roll(-1)
    float w = __powf(wt, gamma);
    wb[(size_t)bh * L_ + l] = w;

    red[threadIdx.x] = w;
    __syncthreads();
    for (int off = 128; off >= 1; off >>= 1) {
        if (threadIdx.x < off) red[threadIdx.x] += red[threadIdx.x + off];
        __syncthreads();
    }
    if (threadIdx.x == 0) psw[bh * NBLK + sub] = red[0];
}

// ---------------------------------------------------------------------------
// fused read-vector + erase/add memory update (single read+write pass of mem)
// grid = B*NBLK blocks of 256 (8 waves x 32 rows); lane owns 2 columns
// ---------------------------------------------------------------------------
__global__ __launch_bounds__(256)
void ntm_read_update(float* __restrict__ ws)
{
    __shared__ float rpart[8][64];
    float* mem        = ws + OFF_MEM;
    const float* co   = ws + OFF_CO;
    const float* wb   = ws + OFF_WB;
    const float* winv = ws + OFF_WSUM;
    float* rws = ws + OFF_RWS;
    float* wws = ws + OFF_WWS;
    float* psr = ws + OFF_PSR;

    int blk = blockIdx.x;
    int b = blk / NBLK, lb = blk % NBLK;
    int tid = threadIdx.x, wv = tid >> 5, lane = tid & 31;
    float winv_r = winv[b * 2 + 0], winv_w = winv[b * 2 + 1];
    int c0 = 2 * lane;
    float ex = sigmoid_f(co[b * COUT_ + 204 + c0]);       // erase = sigmoid(wp[-2W:-W])
    float ey = sigmoid_f(co[b * COUT_ + 204 + c0 + 1]);
    float ax = co[b * COUT_ + 268 + c0];                  // add   = wp[-W:]
    float ay = co[b * COUT_ + 268 + c0 + 1];

    float raccx = 0.f, raccy = 0.f;
    int lbase = lb * LCHUNK + wv * 32;
    for (int r = 0; r < 32; ++r) {
        int l = lbase + r;
        float rwl = wb[(size_t)(b * 2 + 0) * L_ + l] * winv_r;
        float wwl = wb[(size_t)(b * 2 + 1) * L_ + l] * winv_w;
        float* mp = &mem[((size_t)(b * L_ + l)) * W_ + c0];
        __builtin_prefetch(&mem[((size_t)(b * L_ + l + 8)) * W_ + c0], 0, 1);
        float2 m = *(const float2*)mp;
        raccx += rwl * m.x;
        raccy += rwl * m.y;
        float2 nm;
        nm.x = m.x * (1.f - wwl * ex) + wwl * ax;
        nm.y = m.y * (1.f - wwl * ey) + wwl * ay;
        *(float2*)mp = nm;
        if (lane == 0) { rws[b * L_ + l] = rwl; wws[b * L_ + l] = wwl; }
    }
    rpart[wv][c0] = raccx;
    rpart[wv][c0 + 1] = raccy;
    __syncthreads();
    if (tid < 64) {
        float s = 0.f;
        for (int w = 0; w < 8; ++w) s += rpart[w][tid];
        psr[(size_t)(b * NBLK + lb) * W_ + tid] = s;
    }
}

// ---------------------------------------------------------------------------
__global__ __launch_bounds__(64)
void ntm_reduce_r(float* __restrict__ ws)
{
    const float* psr = ws + OFF_PSR;
    float* rst = ws + OFF_RST;
    int b = blockIdx.x, j = threadIdx.x;
    float s = 0.f;
    for (int lb = 0; lb < NBLK; ++lb) s += psr[(size_t)(b * NBLK + lb) * W_ + j];
    rst[b * W_ + j] = s;
}

// ---------------------------------------------------------------------------
extern "C" void kernel_launch(void* const* d_in, const int* in_sizes, int n_in,
                              void* d_out, int out_size, void* d_ws, size_t ws_size,
                              hipStream_t stream)
{
    const float* inputs = (const float*)d_in[0];   // (T,B,IN) f32
    const float* Wc     = (const float*)d_in[1];   // (192,332) f32
    const float* bc     = (const float*)d_in[2];   // (332,) f32
    float* out = (float*)d_out;                    // (T,B,OUT) f32
    float* ws  = (float*)d_ws;
    if (ws_size < (size_t)WS_NEED_FLOATS * sizeof(float)) return;  // ~18.5 MB needed

    ntm_init<<<SZ_MEM / (4 * 256), 256, 0, stream>>>(ws);
    for (int t = 0; t < T_; ++t) {
        ntm_controller<<<1, 768, 0, stream>>>(inputs + (size_t)t * B_ * IN_,
                                              Wc, bc, ws, out, t);
        ntm_score<<<B_ * NBLK, 256, 0, stream>>>(ws);
        ntm_reduce32<<<1, 32, 0, stream>>>(ws + OFF_PSU, ws + OFF_USUM);
        ntm_shift<<<B_ * 2 * NBLK, 256, 0, stream>>>(ws);
        ntm_reduce32<<<1, 32, 0, stream>>>(ws + OFF_PSW, ws + OFF_WSUM);
        ntm_read_update<<<B_ * NBLK, 256, 0, stream>>>(ws);
        ntm_reduce_r<<<B_, 64, 0, stream>>>(ws);
    }
}